// MultiModalMILModel_1382979470023
// MI455X (gfx1250) — compile-verified
//
#include <hip/hip_runtime.h>
#include <math.h>

#define NPTS 4096
#define DM   256
#define NH   4
#define HD   64
#define KNN  9          // k+1 nearest including self
#define NBR_SLOTS 10    // 9 knn + optional forced-self slot
#define MILH 128

typedef __attribute__((ext_vector_type(2))) float v2f;
typedef __attribute__((ext_vector_type(8))) float v8f;

__device__ __forceinline__ float gelu_exact(float x) {
    return 0.5f * x * (1.0f + erff(x * 0.70710678118654752440f));
}

// ---------------- LayerNorm over a row of 256, optional GELU ----------------
__global__ void ln_kernel(const float* __restrict__ in, const float* __restrict__ g,
                          const float* __restrict__ b, float* __restrict__ out,
                          int ldo, int ooff, int do_gelu)
{
    __shared__ float red[DM];
    int row = blockIdx.x;
    int tid = threadIdx.x;
    float x = in[(size_t)row * DM + tid];
    red[tid] = x; __syncthreads();
    #pragma unroll
    for (int s = DM / 2; s > 0; s >>= 1) { if (tid < s) red[tid] += red[tid + s]; __syncthreads(); }
    float mean = red[0] * (1.0f / DM);
    __syncthreads();
    float d = x - mean;
    red[tid] = d * d; __syncthreads();
    #pragma unroll
    for (int s = DM / 2; s > 0; s >>= 1) { if (tid < s) red[tid] += red[tid + s]; __syncthreads(); }
    float var = red[0] * (1.0f / DM);
    float y = d * rsqrtf(var + 1e-5f) * g[tid] + b[tid];
    if (do_gelu) y = gelu_exact(y);
    out[(size_t)row * ldo + ooff + tid] = y;
}

// ---------------- weight pack: fragment-major layout for b128 B-loads ----------------
// P index: (((s*TN + tn)*32 + kg*16 + ml) * 16) + t*2 + j
//   maps to W[(s*4 + 2*kg + j) * Nn + tn*128 + t*16 + ml],  TN = Nn/128
__global__ void pack_w_kernel(const float* __restrict__ W, float* __restrict__ P, int K, int Nn)
{
    int e = blockIdx.x * blockDim.x + threadIdx.x;
    if (e >= K * Nn) return;
    int j    = e & 1;
    int t    = (e >> 1) & 7;
    int ml   = (e >> 4) & 15;
    int kg   = (e >> 8) & 1;
    int tile = e >> 9;
    int TN   = Nn >> 7;
    int tn   = tile % TN;
    int s    = tile / TN;
    int row  = s * 4 + 2 * kg + j;
    int col  = tn * 128 + t * 16 + ml;
    P[e] = W[(size_t)row * Nn + col];
}

// ---------------- fp32 WMMA GEMM: C[M,Nn] = A[M,K] @ W[K,Nn] + bias ----------------
// epilogue: 0=none, 1=GELU, 2=add resid, 3=tanh, 4=sigmoid
// One wave computes a 16x128 tile (8 N-subtiles sharing one A fragment),
// B comes from the packed layout (4x b128 per K-step), K-loop software-pipelined.
__global__ void wmma_gemm_kernel(const float* __restrict__ A, const float* __restrict__ Bpk,
                                 const float* __restrict__ bias, const float* __restrict__ resid,
                                 float* __restrict__ C, int M, int K, int Nn, int ep)
{
    int wavesPerBlock = blockDim.x >> 5;
    int wave = blockIdx.x * wavesPerBlock + (threadIdx.x >> 5);
    int tilesN = Nn >> 7;
    int tm = wave / tilesN;
    int tn = wave % tilesN;
    int rowBase = tm << 4;
    int colBase = tn << 7;
    if (rowBase >= M) return;      // uniform per wave -> EXEC all-ones inside WMMA

    int lane = threadIdx.x & 31;
    int ml   = lane & 15;          // M index (A/C) or N index (B)
    int kg   = lane >> 4;          // K half-group

    const float* Arow = A + (size_t)(rowBase + ml) * K + 2 * kg;
    const float4* Bp  = (const float4*)(Bpk + (((size_t)tn * 32) + kg * 16 + ml) * 16);
    const size_t bStride4 = (size_t)tilesN * 128;   // float4s per K-step

    int steps = K >> 2;
    v8f acc[8] = {};

    v2f a_cur = *(const v2f*)(Arow);
    float4 b_cur[4];
    #pragma unroll
    for (int i = 0; i < 4; i++) b_cur[i] = Bp[i];

    for (int s = 0; s < steps; ++s) {
        bool last = (s == steps - 1);
        v2f a_nxt = {};
        float4 b_nxt[4] = {};
        if (!last) {                              // prefetch next step's fragments
            a_nxt = *(const v2f*)(Arow + (s + 1) * 4);
            const float4* Bn = Bp + (size_t)(s + 1) * bStride4;
            #pragma unroll
            for (int i = 0; i < 4; i++) b_nxt[i] = Bn[i];
        }
        #pragma unroll
        for (int t = 0; t < 8; ++t) {
            float4 qv = b_cur[t >> 1];
            v2f b;
            if (t & 1) { b.x = qv.z; b.y = qv.w; } else { b.x = qv.x; b.y = qv.y; }
            acc[t] = __builtin_amdgcn_wmma_f32_16x16x4_f32(
                false, a_cur, false, b, (short)0, acc[t], false, false);
        }
        a_cur = a_nxt;
        #pragma unroll
        for (int i = 0; i < 4; i++) b_cur[i] = b_nxt[i];
    }

    #pragma unroll
    for (int t = 0; t < 8; t++) {
        int n = colBase + t * 16 + ml;
        float bv = bias[n];
        #pragma unroll
        for (int r = 0; r < 8; r++) {
            int m = rowBase + r + 8 * kg;     // C/D: vgpr r, lane group -> m = r + 8*g
            float val = acc[t][r] + bv;
            if (ep == 1)      val = gelu_exact(val);
            else if (ep == 2) val += resid[(size_t)m * Nn + n];
            else if (ep == 3) val = tanhf(val);
            else if (ep == 4) val = 1.0f / (1.0f + expf(-val));
            C[(size_t)m * Nn + n] = val;
        }
    }
}

// ---------------- kNN: 9 nearest (incl self) per row, tie-break by lower index ----------------
__global__ void knn_kernel(const float* __restrict__ coords, int* __restrict__ nbr)
{
    __shared__ float sd[32 * KNN];
    __shared__ int   si[32 * KNN];
    int row  = blockIdx.x;
    int lane = threadIdx.x;
    float cx = coords[row * 2 + 0], cy = coords[row * 2 + 1];
    float bd[KNN]; int bi[KNN];
    #pragma unroll
    for (int t = 0; t < KNN; t++) { bd[t] = 3.0e38f; bi[t] = 0x7fffffff; }
    for (int j = lane; j < NPTS; j += 32) {
        float dx = cx - coords[j * 2 + 0];
        float dy = cy - coords[j * 2 + 1];
        float d2 = dx * dx + dy * dy;
        if (d2 < bd[KNN - 1] || (d2 == bd[KNN - 1] && j < bi[KNN - 1])) {
            int p = KNN - 1;
            while (p > 0 && (d2 < bd[p - 1] || (d2 == bd[p - 1] && j < bi[p - 1]))) {
                bd[p] = bd[p - 1]; bi[p] = bi[p - 1]; --p;
            }
            bd[p] = d2; bi[p] = j;
        }
    }
    for (int t = 0; t < KNN; t++) { sd[lane * KNN + t] = bd[t]; si[lane * KNN + t] = bi[t]; }
    __syncthreads();
    if (lane == 0) {
        bool self_found = false;
        for (int t = 0; t < KNN; t++) {
            float bv = 3.1e38f; int bj = 0x7fffffff; int best = 0;
            for (int c = 0; c < 32 * KNN; c++) {
                float dc = sd[c]; int jc = si[c];
                if (dc < bv || (dc == bv && jc < bj)) { bv = dc; bj = jc; best = c; }
            }
            sd[best] = 3.2e38f;
            nbr[row * NBR_SLOTS + t] = bj;
            if (bj == row) self_found = true;
        }
        nbr[row * NBR_SLOTS + KNN] = self_found ? -1 : row;
    }
}

// ---------------- sparse kNN-masked attention: one wave per (row, head) ----------------
__global__ void attn_kernel(const float* __restrict__ q, const float* __restrict__ kk,
                            const float* __restrict__ v, const int* __restrict__ nbr,
                            float* __restrict__ out)
{
    int wid = blockIdx.x * (blockDim.x >> 5) + (threadIdx.x >> 5);
    int row = wid >> 2;
    int h   = wid & 3;
    if (row >= NPTS) return;
    int lane = threadIdx.x & 31;
    const float* qp = q + (size_t)row * DM + h * HD;
    float q0 = qp[lane], q1 = qp[lane + 32];
    float s[NBR_SLOTS]; int id[NBR_SLOTS];
    float smax = -3.0e38f;
    #pragma unroll
    for (int t = 0; t < NBR_SLOTS; t++) {
        int j = nbr[row * NBR_SLOTS + t];
        id[t] = j;
        if (j < 0) { s[t] = 0.0f; continue; }
        const float* kp = kk + (size_t)j * DM + h * HD;
        float p = q0 * kp[lane] + q1 * kp[lane + 32];
        #pragma unroll
        for (int off = 16; off > 0; off >>= 1) p += __shfl_xor(p, off, 32);
        p *= 0.125f;                               // 1/sqrt(64)
        s[t] = p;
        smax = fmaxf(smax, p);
    }
    float ssum = 0.0f;
    #pragma unroll
    for (int t = 0; t < NBR_SLOTS; t++) {
        if (id[t] >= 0) { s[t] = expf(s[t] - smax); ssum += s[t]; } else s[t] = 0.0f;
    }
    float inv = 1.0f / ssum;
    float a0 = 0.0f, a1 = 0.0f;
    #pragma unroll
    for (int t = 0; t < NBR_SLOTS; t++) {
        if (id[t] < 0) continue;
        const float* vp = v + (size_t)id[t] * DM + h * HD;
        float wgt = s[t] * inv;
        a0 += wgt * vp[lane];
        a1 += wgt * vp[lane + 32];
    }
    out[(size_t)row * DM + h * HD + lane]      = a0;
    out[(size_t)row * DM + h * HD + lane + 32] = a1;
}

// ---------------- MIL gated score: A[n] = sum_j tanh*sigmoid*wW + wb ----------------
__global__ void mil_score_kernel(const float* __restrict__ tb, const float* __restrict__ ub,
                                 const float* __restrict__ wW, const float* __restrict__ wb,
                                 float* __restrict__ A)
{
    __shared__ float red[MILH];
    int row = blockIdx.x, tid = threadIdx.x;
    red[tid] = tb[(size_t)row * MILH + tid] * ub[(size_t)row * MILH + tid] * wW[tid];
    __syncthreads();
    #pragma unroll
    for (int s = MILH / 2; s > 0; s >>= 1) { if (tid < s) red[tid] += red[tid + s]; __syncthreads(); }
    if (tid == 0) A[row] = red[0] + wb[0];
}

// ---------------- softmax over N=4096 scores, single block ----------------
__global__ void softmax_n_kernel(const float* __restrict__ A, float* __restrict__ Wgt)
{
    __shared__ float red[1024];
    int tid = threadIdx.x;
    float m = -3.0e38f;
    for (int n = tid; n < NPTS; n += 1024) m = fmaxf(m, A[n]);
    red[tid] = m; __syncthreads();
    #pragma unroll
    for (int s = 512; s > 0; s >>= 1) { if (tid < s) red[tid] = fmaxf(red[tid], red[tid + s]); __syncthreads(); }
    float mx = red[0]; __syncthreads();
    float se = 0.0f;
    for (int n = tid; n < NPTS; n += 1024) se += expf(A[n] - mx);
    red[tid] = se; __syncthreads();
    #pragma unroll
    for (int s = 512; s > 0; s >>= 1) { if (tid < s) red[tid] += red[tid + s]; __syncthreads(); }
    float inv = 1.0f / red[0];
    for (int n = tid; n < NPTS; n += 1024) Wgt[n] = expf(A[n] - mx) * inv;
}

// ---------------- weighted pooling + classifier head ----------------
__global__ void final_kernel(const float* __restrict__ Wgt, const float* __restrict__ x2,
                             const float* __restrict__ clsW, const float* __restrict__ clsB,
                             float* __restrict__ out)
{
    __shared__ float wsi[DM];
    int tid = threadIdx.x;
    float acc = 0.0f;
    for (int n = 0; n < NPTS; n++) acc += Wgt[n] * x2[(size_t)n * DM + tid];
    wsi[tid] = acc; __syncthreads();
    if (tid < 2) {
        float l = clsB[tid];
        for (int d = 0; d < DM; d++) l += wsi[d] * clsW[d * 2 + tid];
        out[tid] = l;
    }
}

extern "C" void kernel_launch(void* const* d_in, const int* in_sizes, int n_in,
                              void* d_out, int out_size, void* d_ws, size_t ws_size,
                              hipStream_t stream)
{
    const float* img      = (const float*)d_in[0];
    const float* stf      = (const float*)d_in[1];
    const float* coords   = (const float*)d_in[2];
    const float* pn_img_g = (const float*)d_in[3];
    const float* pn_img_b = (const float*)d_in[4];
    const float* pn_st_g  = (const float*)d_in[5];
    const float* pn_st_b  = (const float*)d_in[6];
    const float* fuse_W   = (const float*)d_in[7];
    const float* fuse_b   = (const float*)d_in[8];
    const float* fuse_ln_g= (const float*)d_in[9];
    const float* fuse_ln_b= (const float*)d_in[10];
    const float* sa_n1_g  = (const float*)d_in[11];
    const float* sa_n1_b  = (const float*)d_in[12];
    const float* sa_qW    = (const float*)d_in[13];
    const float* sa_qb    = (const float*)d_in[14];
    const float* sa_kW    = (const float*)d_in[15];
    const float* sa_kb    = (const float*)d_in[16];
    const float* sa_vW    = (const float*)d_in[17];
    const float* sa_vb    = (const float*)d_in[18];
    const float* sa_oW    = (const float*)d_in[19];
    const float* sa_ob    = (const float*)d_in[20];
    const float* sa_n2_g  = (const float*)d_in[21];
    const float* sa_n2_b  = (const float*)d_in[22];
    const float* sa_f1W   = (const float*)d_in[23];
    const float* sa_f1b   = (const float*)d_in[24];
    const float* sa_f2W   = (const float*)d_in[25];
    const float* sa_f2b   = (const float*)d_in[26];
    const float* mil_VW   = (const float*)d_in[27];
    const float* mil_Vb   = (const float*)d_in[28];
    const float* mil_UW   = (const float*)d_in[29];
    const float* mil_Ub   = (const float*)d_in[30];
    const float* mil_wW   = (const float*)d_in[31];
    const float* mil_wb   = (const float*)d_in[32];
    const float* cls_W    = (const float*)d_in[33];
    const float* cls_b    = (const float*)d_in[34];

    float* w = (float*)d_ws;
    const size_t NF = (size_t)NPTS;
    float* XCAT  = w;                    // N*512  LN'd concat
    float* Y     = XCAT + NF * 512;      // N*256  fusion pre-LN
    float* FUSED = Y    + NF * 256;      // N*256
    float* XN    = FUSED+ NF * 256;      // N*256  (reused: H, then T=tanh N*128)
    float* Q     = XN   + NF * 256;      // N*256  (reused: X)
    float* Kb    = Q    + NF * 256;      // N*256  (reused: X2)
    float* V     = Kb   + NF * 256;      // N*256  (reused: A scores + softmax weights)
    float* AO    = V    + NF * 256;      // N*256  (reused: U=sigmoid N*128)
    float* MID   = AO   + NF * 256;      // N*1024 FFN mid
    int*   NBR   = (int*)(MID + NF * 1024);   // N*10 ints
    float* PK    = (float*)(NBR + NPTS * NBR_SLOTS);
    float* PK_FUSE = PK;                      // 512*256
    float* PK_Q    = PK_FUSE + 512 * 256;     // 256*256
    float* PK_K    = PK_Q    + 256 * 256;
    float* PK_V    = PK_K    + 256 * 256;
    float* PK_O    = PK_V    + 256 * 256;
    float* PK_F1   = PK_O    + 256 * 256;     // 256*1024
    float* PK_F2   = PK_F1   + 256 * 1024;    // 1024*256
    float* PK_VW   = PK_F2   + 1024 * 256;    // 256*128
    float* PK_UW   = PK_VW   + 256 * 128;

    auto pack = [&](const float* Wm, float* Pm, int K, int Nn) {
        int total = K * Nn;
        pack_w_kernel<<<(total + 255) / 256, 256, 0, stream>>>(Wm, Pm, K, Nn);
    };
    auto gemm = [&](const float* A, const float* Pm, const float* bias,
                    const float* resid, float* C, int M, int K, int Nn, int ep) {
        int tiles  = (M / 16) * (Nn / 128);
        int blocks = (tiles + 3) / 4;      // 4 waves (128 threads) per block
        wmma_gemm_kernel<<<blocks, 128, 0, stream>>>(A, Pm, bias, resid, C, M, K, Nn, ep);
    };

    // Pack all weights into fragment-major layout (cheap; overlaps with LN work)
    pack(fuse_W, PK_FUSE, 512, 256);
    pack(sa_qW,  PK_Q,    256, 256);
    pack(sa_kW,  PK_K,    256, 256);
    pack(sa_vW,  PK_V,    256, 256);
    pack(sa_oW,  PK_O,    256, 256);
    pack(sa_f1W, PK_F1,   256, 1024);
    pack(sa_f2W, PK_F2,   1024, 256);
    pack(mil_VW, PK_VW,   256, 128);
    pack(mil_UW, PK_UW,   256, 128);

    // Fusion layer
    ln_kernel<<<NPTS, DM, 0, stream>>>(img, pn_img_g, pn_img_b, XCAT, 512, 0,   0);
    ln_kernel<<<NPTS, DM, 0, stream>>>(stf, pn_st_g,  pn_st_b,  XCAT, 512, 256, 0);
    gemm(XCAT, PK_FUSE, fuse_b, nullptr, Y, NPTS, 512, 256, 0);
    ln_kernel<<<NPTS, DM, 0, stream>>>(Y, fuse_ln_g, fuse_ln_b, FUSED, 256, 0, 1);

    // Spatial attention
    ln_kernel<<<NPTS, DM, 0, stream>>>(FUSED, sa_n1_g, sa_n1_b, XN, 256, 0, 0);
    gemm(XN, PK_Q, sa_qb, nullptr, Q,  NPTS, 256, 256, 0);
    gemm(XN, PK_K, sa_kb, nullptr, Kb, NPTS, 256, 256, 0);
    gemm(XN, PK_V, sa_vb, nullptr, V,  NPTS, 256, 256, 0);
    knn_kernel<<<NPTS, 32, 0, stream>>>(coords, NBR);
    attn_kernel<<<(NPTS * NH) / 8, 256, 0, stream>>>(Q, Kb, V, NBR, AO);
    gemm(AO, PK_O, sa_ob, FUSED, Q /* X */, NPTS, 256, 256, 2);

    // FFN
    ln_kernel<<<NPTS, DM, 0, stream>>>(Q /* X */, sa_n2_g, sa_n2_b, XN /* H */, 256, 0, 0);
    gemm(XN, PK_F1, sa_f1b, nullptr, MID, NPTS, 256, 1024, 1);
    gemm(MID, PK_F2, sa_f2b, Q /* resid X */, Kb /* X2 */, NPTS, 1024, 256, 2);

    // MIL attention pooling
    gemm(Kb, PK_VW, mil_Vb, nullptr, XN /* T */, NPTS, 256, 128, 3);
    gemm(Kb, PK_UW, mil_Ub, nullptr, AO /* U */, NPTS, 256, 128, 4);
    mil_score_kernel<<<NPTS, MILH, 0, stream>>>(XN, AO, mil_wW, mil_wb, V /* A */);
    softmax_n_kernel<<<1, 1024, 0, stream>>>(V, V + NPTS /* weights */);
    final_kernel<<<1, DM, 0, stream>>>(V + NPTS, Kb, cls_W, cls_b, (float*)d_out);
}